// CrossAttention_61873298866679
// MI455X (gfx1250) — compile-verified
//
#include <hip/hip_runtime.h>

// ---------------- problem constants ----------------
#define BB 64
#define NN 196
#define DD 768
#define HH 12
#define HD 64
#define NPAD 208          // 13 * 16
#define QTILES 13         // ceil(196/16)
#define MTILES 784        // 12544 / 16
#define NT64 (DD / 64)    // 12
#define ASTRIDE 772       // 768 + 4 pad floats: 16B-aligned rows, bank-conflict free

typedef __attribute__((ext_vector_type(2))) float v2f;
typedef __attribute__((ext_vector_type(4))) float v4f;
typedef __attribute__((ext_vector_type(8))) float v8f;

#define WMMA_F32(a, b, c) \
    __builtin_amdgcn_wmma_f32_16x16x4_f32(false, (a), false, (b), (short)0, (c), false, false)

#if __has_builtin(__builtin_amdgcn_global_load_async_to_lds_b128) && \
    __has_builtin(__builtin_amdgcn_s_wait_asynccnt)
#define HAVE_ASYNC_LDS 1
// exact parameter types per hipcc diagnostic: int4 (vector_size 16) pointers,
// addrspace(1) for the global source, addrspace(3) for the LDS destination.
typedef __attribute__((vector_size(16))) int vsi4;
typedef __attribute__((address_space(1))) vsi4 g_vsi4;
typedef __attribute__((address_space(3))) vsi4 l_vsi4;
#else
#define HAVE_ASYNC_LDS 0
#endif

// ---------------------------------------------------------------------------
// Stage 1: zero only the 12 padding rows (tokens 196..207) of Q/K/V head
// buffers [B][H][NPAD][HD] — replaces a 123 MB memset with ~7 MB of writes.
// ---------------------------------------------------------------------------
__global__ __launch_bounds__(256) void zero_pad_kernel(float* __restrict__ ws) {
  const size_t QHN = (size_t)BB * HH * NPAD * HD;
  size_t idx = (size_t)blockIdx.x * 256 + threadIdx.x;  // < 3*BB*HH*12*HD
  const size_t per_mat = (size_t)BB * HH * 12 * HD;
  int mat = (int)(idx / per_mat);
  size_t rem = idx % per_mat;
  int hd = (int)(rem % HD);
  size_t t = rem / HD;
  int row = NN + (int)(t % 12);
  int bh = (int)(t / 12);  // 0..767
  ws[mat * QHN + ((size_t)bh * NPAD + row) * HD + hd] = 0.0f;
}

// ---------------------------------------------------------------------------
// Stage 2: relative-position bias MLP, fused with key-padding mask.
// bias[b,q,k] = kpad ? -inf : ((is_overlap && !qpad) ? enc(qpos-kpos) : 0)
// Stored [B][NPAD][NPAD] (only [0..195][0..195] written / read).
// ---------------------------------------------------------------------------
__global__ __launch_bounds__(256) void bias_kernel(
    const float* __restrict__ qpos, const float* __restrict__ kpos,
    const int* __restrict__ qpad, const int* __restrict__ kpad,
    const int* __restrict__ is_ovl,
    const float* __restrict__ W1, const float* __restrict__ b1,
    const float* __restrict__ W2, const float* __restrict__ b2,
    float* __restrict__ biasbuf) {
  size_t idx = (size_t)blockIdx.x * 256 + threadIdx.x;
  if (idx >= (size_t)BB * NN * NN) return;
  int k = (int)(idx % NN);
  size_t r = idx / NN;
  int q = (int)(r % NN);
  int b = (int)(r / NN);

  float out;
  if (kpad[b * NN + k]) {
    out = -__builtin_inff();
  } else if (is_ovl[b] && !qpad[b * NN + q]) {
    float rx = qpos[((size_t)b * NN + q) * 3 + 0] - kpos[((size_t)b * NN + k) * 3 + 0];
    float ry = qpos[((size_t)b * NN + q) * 3 + 1] - kpos[((size_t)b * NN + k) * 3 + 1];
    float rz = qpos[((size_t)b * NN + q) * 3 + 2] - kpos[((size_t)b * NN + k) * 3 + 2];
    float acc = b2[0];
#pragma unroll
    for (int e = 0; e < 32; ++e) {
      float h = fmaf(rx, W1[e], fmaf(ry, W1[32 + e], fmaf(rz, W1[64 + e], b1[e])));
      acc = fmaf(fmaxf(h, 0.0f), W2[e], acc);
    }
    out = acc;
  } else {
    out = 0.0f;
  }
  biasbuf[((size_t)b * NPAD + q) * NPAD + k] = out;
}

// ---------------------------------------------------------------------------
// Stages 3 & 5: fp32 WMMA GEMM, C[12544,768] = A[12544,768] * W[768,768].
// Block = 12 waves = one 16-row M-tile; wave w owns 64 output columns.
// The 16x768 A slab is staged once into LDS with async global->LDS copies
// (ASYNCcnt), then A fragments come from LDS (ds_load_b64, bank-conflict-free).
// mode 0: scatter into padded head layout [B][H][NPAD][HD]   (Q/K/V proj)
// mode 1: out[row*768+col] = acc + bias[col]                 (final proj)
// ---------------------------------------------------------------------------
__global__ __launch_bounds__(384) void gemm_proj(
    const float* __restrict__ A, const float* __restrict__ W,
    const float* __restrict__ bias, float* __restrict__ out, int mode) {
  __shared__ float Abuf[16][ASTRIDE];  // 49.4 KB

  int mtile = blockIdx.x;          // 0..783
  int nt = threadIdx.x >> 5;       // wave id = 64-col tile (0..11)
  int l = threadIdx.x & 31;
  int h = l >> 4;                  // lane half: selects k-pair / row-half
  int n16 = l & 15;                // lane within 16

  // ---- stage A slab [16][768] into LDS, coalesced ----
  {
    const float* asrc = A + (size_t)mtile * 16 * DD;
#pragma unroll
    for (int i = 0; i < 8; ++i) {
      int idx = (int)threadIdx.x + i * 384;  // float4 index 0..3071
      int row = idx / 192;
      int c4 = (idx % 192) * 4;
#if HAVE_ASYNC_LDS
      __builtin_amdgcn_global_load_async_to_lds_b128(
          (g_vsi4*)(asrc + (size_t)row * DD + c4),
          (l_vsi4*)&Abuf[row][c4], 0, 0);
#else
      *(v4f*)&Abuf[row][c4] = *(const v4f*)(asrc + (size_t)row * DD + c4);
#endif
    }
#if HAVE_ASYNC_LDS
    __builtin_amdgcn_s_wait_asynccnt(0);
#endif
    __syncthreads();
  }

  v8f acc[4] = {v8f{}, v8f{}, v8f{}, v8f{}};

  for (int k = 0; k < DD; k += 4) {
    // A fragment from LDS: element j -> A[row=n16][k + 2h + j]
    v2f a = *(const v2f*)&Abuf[n16][k + 2 * h];
#pragma unroll
    for (int c4 = 0; c4 < 4; ++c4) {
      int col = nt * 64 + c4 * 16 + n16;
      v2f bfr;
      bfr.x = W[(size_t)(k + 2 * h + 0) * DD + col];
      bfr.y = W[(size_t)(k + 2 * h + 1) * DD + col];
      acc[c4] = WMMA_F32(a, bfr, acc[c4]);
    }
  }

#pragma unroll
  for (int c4 = 0; c4 < 4; ++c4) {
#pragma unroll
    for (int j = 0; j < 8; ++j) {
      int m = j + 8 * h;                 // C row within tile
      int R = mtile * 16 + m;            // global row
      int col = nt * 64 + c4 * 16 + n16; // global col
      if (mode == 0) {
        int bb = R / NN, tok = R % NN;
        int hh = col >> 6, hd = col & 63;
        out[(((size_t)bb * HH + hh) * NPAD + tok) * HD + hd] = acc[c4][j];
      } else {
        out[(size_t)R * DD + col] = acc[c4][j] + bias[col];
      }
    }
  }
}

// ---------------------------------------------------------------------------
// Stage 4: attention. Block = 4 waves, block per (batch, query-tile).
// Each wave processes 3 heads. Scores -> LDS [16][NPAD] -> softmax -> P@V.
// ---------------------------------------------------------------------------
__global__ __launch_bounds__(128) void attention_kernel(
    const float* __restrict__ Qh, const float* __restrict__ Kh,
    const float* __restrict__ Vh, const float* __restrict__ biasbuf,
    float* __restrict__ ctx) {
  __shared__ float S[4][16][NPAD];  // 52 KB

  int w = threadIdx.x >> 5;
  int l = threadIdx.x & 31;
  int b = blockIdx.x / QTILES;
  int qt = blockIdx.x % QTILES;
  int h = l >> 4;
  int n16 = l & 15;
  const float scale = 0.125f;  // 1/sqrt(HD)

  for (int head = w; head < HH; head += 4) {
    const float* Qb = Qh + ((size_t)b * HH + head) * NPAD * HD;
    const float* Kb = Kh + ((size_t)b * HH + head) * NPAD * HD;
    const float* Vb = Vh + ((size_t)b * HH + head) * NPAD * HD;

    // ---- Q fragments: 16 queries x 64 head-dims, 16 k-chunks of 4 ----
    int qrow = qt * 16 + n16;  // < NPAD, padded region is zero
    v2f qf[16];
#pragma unroll
    for (int kc = 0; kc < 16; ++kc)
      qf[kc] = *(const v2f*)(Qb + (size_t)qrow * HD + kc * 4 + 2 * h);

    // ---- scores: S = scale*(Q K^T) + bias, key-padding masked ----
    for (int t = 0; t < QTILES; ++t) {
      int key = t * 16 + n16;
      const float* krow = Kb + (size_t)key * HD;
      v8f acc = v8f{};
#pragma unroll
      for (int kc = 0; kc < 16; ++kc) {
        int hd0 = kc * 4 + 2 * h;
        v2f bfr;
        bfr.x = krow[hd0];
        bfr.y = krow[hd0 + 1];
        acc = WMMA_F32(qf[kc], bfr, acc);
      }
#pragma unroll
      for (int j = 0; j < 8; ++j) {
        int m = j + 8 * h;
        int q = qt * 16 + m;
        float s = acc[j] * scale;
        if (key < NN && q < NN)
          s += biasbuf[((size_t)b * NPAD + q) * NPAD + key];  // holds -inf for kpad
        if (key >= NN) s = -__builtin_inff();
        S[w][m][t * 16 + n16] = s;
      }
    }
    __syncthreads();

    // ---- softmax along each of the 16 rows (208 cols, -inf in padding) ----
    for (int m = 0; m < 16; ++m) {
      float mx = -__builtin_inff();
      for (int c = l; c < NPAD; c += 32) mx = fmaxf(mx, S[w][m][c]);
#pragma unroll
      for (int off = 16; off >= 1; off >>= 1) mx = fmaxf(mx, __shfl_xor(mx, off, 32));
      float sum = 0.0f;
      for (int c = l; c < NPAD; c += 32) {
        float e = __expf(S[w][m][c] - mx);
        S[w][m][c] = e;
        sum += e;
      }
#pragma unroll
      for (int off = 16; off >= 1; off >>= 1) sum += __shfl_xor(sum, off, 32);
      float inv = 1.0f / sum;
      for (int c = l; c < NPAD; c += 32) S[w][m][c] *= inv;
    }
    __syncthreads();

    // ---- out = P @ V : A-fragments re-read from LDS in A-layout ----
    v8f oacc[4] = {v8f{}, v8f{}, v8f{}, v8f{}};
    for (int t = 0; t < NPAD / 4; ++t) {
      int k0 = t * 4 + 2 * h;
      v2f af = *(const v2f*)&S[w][n16][k0];  // ds_load_b64
#pragma unroll
      for (int oc = 0; oc < 4; ++oc) {
        v2f bfr;
        bfr.x = Vb[(size_t)(k0 + 0) * HD + oc * 16 + n16];
        bfr.y = Vb[(size_t)(k0 + 1) * HD + oc * 16 + n16];
        oacc[oc] = WMMA_F32(af, bfr, oacc[oc]);
      }
    }
#pragma unroll
    for (int oc = 0; oc < 4; ++oc) {
#pragma unroll
      for (int j = 0; j < 8; ++j) {
        int m = j + 8 * h;
        int q = qt * 16 + m;
        if (q < NN)
          ctx[((size_t)b * NN + q) * DD + head * HD + oc * 16 + n16] = oacc[oc][j];
      }
    }
    __syncthreads();
  }
}

// ---------------------------------------------------------------------------
extern "C" void kernel_launch(void* const* d_in, const int* in_sizes, int n_in,
                              void* d_out, int out_size, void* d_ws, size_t ws_size,
                              hipStream_t stream) {
  const float* query = (const float*)d_in[0];
  const float* key   = (const float*)d_in[1];
  const float* value = (const float*)d_in[2];
  const int*   qpad  = (const int*)d_in[3];
  const int*   kpad  = (const int*)d_in[4];
  const float* qpos  = (const float*)d_in[5];
  const float* kpos  = (const float*)d_in[6];
  const int*   isov  = (const int*)d_in[7];
  const float* Wq    = (const float*)d_in[8];
  const float* Wk    = (const float*)d_in[9];
  const float* Wv    = (const float*)d_in[10];
  const float* Wp    = (const float*)d_in[11];
  const float* bp    = (const float*)d_in[12];
  const float* W1    = (const float*)d_in[13];
  const float* b1    = (const float*)d_in[14];
  const float* W2    = (const float*)d_in[15];
  const float* b2    = (const float*)d_in[16];

  float* ws = (float*)d_ws;
  const size_t QHN = (size_t)BB * HH * NPAD * HD;  // 10,223,616 floats
  float* Qh = ws;
  float* Kh = Qh + QHN;
  float* Vh = Kh + QHN;
  float* biasbuf = Vh + QHN;                       // [B][NPAD][NPAD]
  const size_t BIASN = (size_t)BB * NPAD * NPAD;
  float* ctx = biasbuf + BIASN;                    // [B][N][D]

  // stage 1: zero padding rows of Q/K/V head buffers (WMMA tiles read them)
  {
    size_t total = (size_t)3 * BB * HH * 12 * HD;  // 1,769,472 -> 6912 blocks
    zero_pad_kernel<<<(unsigned)(total / 256), 256, 0, stream>>>(ws);
  }

  // stage 2: rel-pos bias + key mask
  {
    size_t total = (size_t)BB * NN * NN;
    bias_kernel<<<(unsigned)((total + 255) / 256), 256, 0, stream>>>(
        qpos, kpos, qpad, kpad, isov, W1, b1, W2, b2, biasbuf);
  }

  // stage 3: Q/K/V projections (async-LDS staged A, WMMA, head-layout scatter)
  gemm_proj<<<MTILES, 384, 0, stream>>>(query, Wq, nullptr, Qh, 0);
  gemm_proj<<<MTILES, 384, 0, stream>>>(key,   Wk, nullptr, Kh, 0);
  gemm_proj<<<MTILES, 384, 0, stream>>>(value, Wv, nullptr, Vh, 0);

  // stage 4: attention
  attention_kernel<<<BB * QTILES, 128, 0, stream>>>(Qh, Kh, Vh, biasbuf, ctx);

  // stage 5: output projection + bias
  gemm_proj<<<MTILES, 384, 0, stream>>>(ctx, Wp, bp, (float*)d_out, 1);
}